// NKSSummaryLoss_17875653886468
// MI455X (gfx1250) — compile-verified
//
#include <hip/hip_runtime.h>

#define BDIM 8192
#define TDIM 128
#define NSPLIT 8

typedef float v2f __attribute__((ext_vector_type(2)));
typedef float v8f __attribute__((ext_vector_type(8)));

// ---------------------------------------------------------------------------
// Kernel 1: per-row pass. One wave32 per row (4 elems/lane, float4 coalesced).
// Produces E[j,t] = exp(-surv[j,t]/sigma), nll[j], coef[j] = e_j*exp(surv_j(d_j)/sigma)
// ---------------------------------------------------------------------------
__global__ __launch_bounds__(256) void k1_rows(const float* __restrict__ hz,
                                               const int* __restrict__ dur,
                                               const int* __restrict__ ev,
                                               float* __restrict__ E,
                                               float* __restrict__ nll,
                                               float* __restrict__ coef) {
  const int wave = threadIdx.x >> 5;
  const int lane = threadIdx.x & 31;
  const int row = blockIdx.x * (blockDim.x >> 5) + wave;

  const float4 hv = reinterpret_cast<const float4*>(hz)[row * (TDIM / 4) + lane];
  float h[4] = {hv.x, hv.y, hv.z, hv.w};
  float l1mh[4], slog[4];
#pragma unroll
  for (int k = 0; k < 4; ++k) {
    l1mh[k] = fmaxf(log1pf(-h[k]), -100.0f);        // clipped log(1-h)
    slog[k] = logf(1.0f - h[k] + 1e-12f);           // surv log-increment
  }
  // lane-local inclusive prefix of slog (t = lane*4 + k)
  float p[4];
  p[0] = slog[0]; p[1] = p[0] + slog[1]; p[2] = p[1] + slog[2]; p[3] = p[2] + slog[3];
  // wave32 exclusive scan of lane totals
  float tot = p[3];
  float incl = tot;
#pragma unroll
  for (int d = 1; d < 32; d <<= 1) {
    float up = __shfl_up(incl, d, 32);
    if (lane >= d) incl += up;
  }
  const float excl = incl - tot;

  const int idx = dur[row];
  const int evi = ev[row];
  float c = 0.0f, co = 0.0f;
  float ee[4];
#pragma unroll
  for (int k = 0; k < 4; ++k) {
    const float surv = expf(excl + p[k]);            // cumsum -> surv
    ee[k] = expf(-10.0f * surv);                      // E = exp(-surv/sigma)
    const int t = lane * 4 + k;
    if (t < idx) {
      c -= l1mh[k];
    } else if (t == idx) {
      const float lh = fmaxf(logf(h[k]), -100.0f);
      c -= (evi != 0) ? lh : l1mh[k];
      co = (evi != 0) ? expf(10.0f * surv) : 0.0f;    // e_i * exp(surv_i(d_i)/sigma)
    }
  }
  float4 Ev; Ev.x = ee[0]; Ev.y = ee[1]; Ev.z = ee[2]; Ev.w = ee[3];
  reinterpret_cast<float4*>(E)[row * (TDIM / 4) + lane] = Ev;

#pragma unroll
  for (int off = 16; off > 0; off >>= 1) {
    c += __shfl_down(c, off, 32);
    co += __shfl_down(co, off, 32);
  }
  if (lane == 0) { nll[row] = c; coef[row] = co; }
}

// ---------------------------------------------------------------------------
// Kernel 2: bucket reduction C = onehot(d)^T @ E  (and event-masked C0) via
// V_WMMA_F32_16X16X4_F32. Block = 8 waves, wave w owns u-tile w; grid.x = 8
// t-tiles, grid.y = 8 K-splits (partials summed deterministically in k3).
// A-matrix (16x4 f32): lanes 0-15 -> M rows, v0/v1 hold K = 2*half + {0,1}.
// ---------------------------------------------------------------------------
__global__ __launch_bounds__(256) void k2_bucket_gemm(const float* __restrict__ E,
                                                      const int* __restrict__ dur,
                                                      const int* __restrict__ ev,
                                                      float* __restrict__ Cp,
                                                      float* __restrict__ C0p) {
  const int wave = threadIdx.x >> 5;
  const int lane = threadIdx.x & 31;
  const int half = lane >> 4;
  const int m = lane & 15;
  const int u0 = wave * 16;
  const int t0 = blockIdx.x * 16;
  const int jbase = blockIdx.y * (BDIM / NSPLIT);

  v8f acc = 0.0f;
  v8f acc0 = 0.0f;

  for (int j0 = jbase; j0 < jbase + (BDIM / NSPLIT); j0 += 4) {
    const int kb = j0 + half * 2;
    const int d0 = dur[kb], d1 = dur[kb + 1];
    const int e0 = ev[kb], e1 = ev[kb + 1];
    v2f a, a0, b;
    a.x = (d0 == u0 + m) ? 1.0f : 0.0f;                 // one-hot bucket matrix
    a.y = (d1 == u0 + m) ? 1.0f : 0.0f;
    a0.x = (d0 == u0 + m && e0 == 0) ? 1.0f : 0.0f;     // event==0 masked
    a0.y = (d1 == u0 + m && e1 == 0) ? 1.0f : 0.0f;
    b.x = E[kb * TDIM + t0 + m];
    b.y = E[(kb + 1) * TDIM + t0 + m];
    acc = __builtin_amdgcn_wmma_f32_16x16x4_f32(false, a, false, b, (short)0, acc,
                                                false, false);
    acc0 = __builtin_amdgcn_wmma_f32_16x16x4_f32(false, a0, false, b, (short)0, acc0,
                                                 false, false);
  }

  float* __restrict__ cpo = Cp + blockIdx.y * (TDIM * TDIM);
  float* __restrict__ c0o = C0p + blockIdx.y * (TDIM * TDIM);
#pragma unroll
  for (int g = 0; g < 8; ++g) {
    const int rowo = u0 + g + 8 * half;   // C/D layout: VGPR g -> M=g / M=g+8
    const int colo = t0 + m;              // N = lane & 15
    cpo[rowo * TDIM + colo] = acc[g];
    c0o[rowo * TDIM + colo] = acc0[g];
  }
}

// ---------------------------------------------------------------------------
// Kernel 3: single block. Sum K-split partials, suffix-sum over u, gather
// diagonal C[d,d], deterministic tree reduction, write scalar loss.
// ---------------------------------------------------------------------------
__global__ __launch_bounds__(256) void k3_finalize(const float* __restrict__ Cp,
                                                   const float* __restrict__ C0p,
                                                   const float* __restrict__ nll,
                                                   const float* __restrict__ coef,
                                                   const int* __restrict__ dur,
                                                   float* __restrict__ ssuf,
                                                   float* __restrict__ c0tot,
                                                   float* __restrict__ out) {
  __shared__ float red[512];
  const int tid = threadIdx.x;

  // combine K-split partials (fixed order -> deterministic)
  for (int n = tid; n < TDIM * TDIM; n += blockDim.x) {
    float s = 0.0f, s0 = 0.0f;
#pragma unroll
    for (int sp = 0; sp < NSPLIT; ++sp) {
      s += Cp[sp * TDIM * TDIM + n];
      s0 += C0p[sp * TDIM * TDIM + n];
    }
    ssuf[n] = s;
    c0tot[n] = s0;
  }
  __syncthreads();

  // suffix sum along u: ssuf[u][t] = sum_{v>u} C[v][t]
  if (tid < TDIM) {
    float run = 0.0f;
    for (int u = TDIM - 1; u >= 0; --u) {
      const float cv = ssuf[u * TDIM + tid];
      ssuf[u * TDIM + tid] = run;
      run += cv;
    }
  }
  __syncthreads();

  float accN = 0.0f, accR = 0.0f;
  for (int i = tid; i < BDIM; i += blockDim.x) {
    accN += nll[i];
    const int di = dur[i];                       // d_i == idx_i -> diagonal gather
    accR += coef[i] * (ssuf[di * (TDIM + 1)] + c0tot[di * (TDIM + 1)]);
  }
  red[tid] = accN;
  red[blockDim.x + tid] = accR;
  __syncthreads();
  for (int s = blockDim.x >> 1; s > 0; s >>= 1) {
    if (tid < s) {
      red[tid] += red[tid + s];
      red[blockDim.x + tid] += red[blockDim.x + tid + s];
    }
    __syncthreads();
  }
  if (tid == 0) {
    const float mn = red[0] / (float)BDIM;
    const float mr = red[blockDim.x] / ((float)BDIM * (float)BDIM);
    out[0] = 0.5f * mn + 0.5f * mr;              // ALPHA = 0.5
  }
}

// ---------------------------------------------------------------------------
extern "C" void kernel_launch(void* const* d_in, const int* in_sizes, int n_in,
                              void* d_out, int out_size, void* d_ws, size_t ws_size,
                              hipStream_t stream) {
  const float* hz = (const float*)d_in[0];   // hazards (B,T) f32
  const int* dur = (const int*)d_in[1];      // idx_durations (B,) i32
  const int* ev = (const int*)d_in[2];       // events (B,) i32
  float* out = (float*)d_out;

  float* ws = (float*)d_ws;
  float* E     = ws;                                     // 1,048,576 f
  float* nll   = ws + 1048576;                           //     8,192 f
  float* coef  = ws + 1056768;                           //     8,192 f
  float* Cp    = ws + 1064960;                           //   131,072 f
  float* C0p   = ws + 1196032;                           //   131,072 f
  float* ssuf  = ws + 1327104;                           //    16,384 f
  float* c0tot = ws + 1343488;                           //    16,384 f  (~5.2 MB total)

  k1_rows<<<dim3(BDIM / 8), dim3(256), 0, stream>>>(hz, dur, ev, E, nll, coef);
  k2_bucket_gemm<<<dim3(TDIM / 16, NSPLIT), dim3(256), 0, stream>>>(E, dur, ev, Cp, C0p);
  k3_finalize<<<dim3(1), dim3(256), 0, stream>>>(Cp, C0p, nll, coef, dur, ssuf, c0tot, out);
}